// ListwiseCrossEntropyLoss_41240275976285
// MI455X (gfx1250) — compile-verified
//
#include <hip/hip_runtime.h>
#include <math.h>

#define BATCH   1024
#define NPOS    10
#define COLS    1000
#define NEGC    990
#define SW      20001                 // stats row stride (NI+1)
#define N_STATS 200030001LL           // (NU+1)*(NI+1)
#define MOM     0.9f
#define EPSV    1e-10f
#define FAIRW   100000.0f

typedef float v2f __attribute__((ext_vector_type(2)));
typedef float v8f __attribute__((ext_vector_type(8)));

// ---- wave32 all-reduce sum via V_WMMA_F32_16X16X4_F32 ----
// A(16x4): lane m holds {x,0} -> A[m,0]=x_m, A[m,2]=x_{m+16}; B = ones(4x16).
// D[m,n] = x_m + x_{m+16} (bcast over n). Sum D vgprs 0..7 => lanes0-15 get
// sum(m=0..7), lanes16-31 get sum(m=8..15); one xor-16 shuffle finishes.
__device__ __forceinline__ float wave_allreduce_sum(float x) {
  v2f a; a.x = x;   a.y = 0.0f;
  v2f b; b.x = 1.0f; b.y = 1.0f;
  v8f c = {};
  c = __builtin_amdgcn_wmma_f32_16x16x4_f32(false, a, false, b,
                                            (short)0, c, false, false);
  float s = c[0] + c[1] + c[2] + c[3] + c[4] + c[5] + c[6] + c[7];
  s += __shfl_xor(s, 16, 32);
  return s;
}

// valid on thread 0 only; safe to call repeatedly with same lds buffer
__device__ __forceinline__ float block_sum(float v, float* lds) {
  float w = wave_allreduce_sum(v);
  if ((threadIdx.x & 31) == 0) lds[threadIdx.x >> 5] = w;
  __syncthreads();
  float t = 0.0f;
  if (threadIdx.x == 0) {
    int nw = blockDim.x >> 5;
    for (int i = 0; i < nw; ++i) t += lds[i];
  }
  __syncthreads();
  return t;
}

__device__ __forceinline__ float block_max(float v, float* lds) {
  for (int o = 16; o; o >>= 1) v = fmaxf(v, __shfl_xor(v, o, 32));
  if ((threadIdx.x & 31) == 0) lds[threadIdx.x >> 5] = v;
  __syncthreads();
  float t = -INFINITY;
  if (threadIdx.x == 0) {
    int nw = blockDim.x >> 5;
    for (int i = 0; i < nw; ++i) t = fmaxf(t, lds[i]);
  }
  __syncthreads();
  return t;
}

// K1: per-row full max (softmax), max over neg cols, min over pos cols
__global__ void k_rowstats(const float* __restrict__ pred,
                           float* __restrict__ rmax, float* __restrict__ cand) {
  int b = blockIdx.x;
  const float* row = pred + (size_t)b * COLS;
  float vmax = -INFINITY, vmaxneg = -INFINITY, vminpos = INFINITY;
  for (int c = threadIdx.x; c < COLS; c += blockDim.x) {
    float x = row[c];
    vmax = fmaxf(vmax, x);
    if (c < NPOS) vminpos = fminf(vminpos, x);
    else          vmaxneg = fmaxf(vmaxneg, x);
  }
  __shared__ float lds[8];
  float m1 = block_max(vmax, lds);
  float m2 = block_max(vmaxneg, lds);
  float m3 = -block_max(-vminpos, lds);
  if (threadIdx.x == 0) { rmax[b] = m1; cand[b] = m2 - m3; }
}

// K1b: global margin max M = max_b cand[b]
__global__ void k_gmax(const float* __restrict__ cand, float* __restrict__ Mout) {
  float v = -INFINITY;
  for (int i = threadIdx.x; i < BATCH; i += blockDim.x) v = fmaxf(v, cand[i]);
  __shared__ float lds[8];
  float m = block_max(v, lds);
  if (threadIdx.x == 0) Mout[0] = m;
}

// K2: per-row S0,S1 (margin exps) + softmax Z, group-diff D -> fairness
__global__ void k_rowsums(const float* __restrict__ pred,
                          const float* __restrict__ aidx,
                          const float* __restrict__ bidx,
                          const float* __restrict__ Mptr,
                          const float* __restrict__ rmaxArr,
                          float* __restrict__ S0, float* __restrict__ S1,
                          float* __restrict__ fair) {
  int b = blockIdx.x;
  const float M  = Mptr[0];
  const float rm = rmaxArr[b];
  const float* row = pred + (size_t)b * COLS;
  const float* ar  = aidx + (size_t)b * COLS;
  const float* br  = bidx + (size_t)b * COLS;
  float s0 = 0.f, s1 = 0.f, z = 0.f, d = 0.f;
  for (int c = threadIdx.x; c < COLS; c += blockDim.x) {
    float x = row[c];
    float u = expf(x - rm);               // softmax numerator
    z += u;
    d += (br[c] - ar[c]) * u;
    if (c >= NPOS) {
      float t = expf(x - M);              // margin exp base
      s0 += t;
      s1 += x * t;
    }
  }
  __shared__ float lds[8];
  float S0b = block_sum(s0, lds);
  float S1b = block_sum(s1, lds);
  float Zb  = block_sum(z,  lds);
  float Db  = block_sum(d,  lds);
  if (threadIdx.x == 0) {
    S0[b] = S0b;
    S1[b] = S1b;
    float diff = Db / Zb;
    fair[b] = diff * diff * FAIRW;
  }
}

// K3: per-(b,p) gather + momentum update + loss contribution (no scatter yet)
__global__ void k_update(const float* __restrict__ pred,
                         const float* __restrict__ stats,
                         const int* __restrict__ uid, const int* __restrict__ iid,
                         const float* __restrict__ S0, const float* __restrict__ S1,
                         float* __restrict__ updated, float* __restrict__ contrib) {
  int idx = blockIdx.x * blockDim.x + threadIdx.x;
  if (idx >= BATCH * NPOS) return;
  int b = idx / NPOS, p = idx - b * NPOS;
  float pos = pred[(size_t)b * COLS + p];
  int u  = uid[b];
  int it = iid[idx];
  float cur = stats[(size_t)u * SW + it];
  float ep   = expf(-pos);
  float s0   = S0[b];
  float mean = ep * s0 * (1.0f / (float)NEGC);
  float upd  = (1.0f - MOM) * cur + MOM * mean;
  updated[idx] = upd;
  contrib[idx] = ep * (S1[b] - pos * s0) / (upd + EPSV);
}

// K4: scatter updated entries into the copied new_stats (runs after memcpy)
__global__ void k_scatter(const int* __restrict__ uid, const int* __restrict__ iid,
                          const float* __restrict__ updated,
                          float* __restrict__ outStats) {
  int idx = blockIdx.x * blockDim.x + threadIdx.x;
  if (idx >= BATCH * NPOS) return;
  int b = idx / NPOS;
  outStats[(size_t)uid[b] * SW + iid[idx]] = updated[idx];
}

// K5: deterministic final reduction -> loss scalar
__global__ void k_final(const float* __restrict__ contrib,
                        const float* __restrict__ fair, float* __restrict__ out) {
  float s = 0.f, f = 0.f;
  for (int i = threadIdx.x; i < BATCH * NPOS; i += blockDim.x) s += contrib[i];
  for (int i = threadIdx.x; i < BATCH;        i += blockDim.x) f += fair[i];
  __shared__ float lds[8];
  float ts = block_sum(s, lds);
  float tf = block_sum(f, lds);
  if (threadIdx.x == 0)
    out[0] = ts * (1.0f / (float)BATCH) + tf * (1.0f / (float)BATCH);
}

extern "C" void kernel_launch(void* const* d_in, const int* in_sizes, int n_in,
                              void* d_out, int out_size, void* d_ws, size_t ws_size,
                              hipStream_t stream) {
  const float* pred  = (const float*)d_in[0];
  const float* stats = (const float*)d_in[1];
  const int*   uid   = (const int*)d_in[2];
  const int*   iid   = (const int*)d_in[3];
  const float* aidx  = (const float*)d_in[4];
  const float* bidx  = (const float*)d_in[5];
  float* out = (float*)d_out;

  float* ws      = (float*)d_ws;
  float* Mcell   = ws;                   // 1
  float* rmax    = ws + 16;              // BATCH
  float* cand    = rmax + BATCH;         // BATCH
  float* S0      = cand + BATCH;         // BATCH
  float* S1      = S0 + BATCH;           // BATCH
  float* fair    = S1 + BATCH;           // BATCH
  float* updated = fair + BATCH;         // BATCH*NPOS
  float* contrib = updated + BATCH*NPOS; // BATCH*NPOS

  // Bulk 800MB pass-through: stats -> new_stats (d_out[1..]); dominates runtime
  // at ~1.6GB / 23.3TB/s. Runs first in stream order; scatter comes after.
  hipMemcpyAsync(out + 1, stats, (size_t)N_STATS * sizeof(float),
                 hipMemcpyDeviceToDevice, stream);

  k_rowstats<<<BATCH, 256, 0, stream>>>(pred, rmax, cand);
  k_gmax   <<<1,     256, 0, stream>>>(cand, Mcell);
  k_rowsums<<<BATCH, 256, 0, stream>>>(pred, aidx, bidx, Mcell, rmax, S0, S1, fair);
  k_update <<<(BATCH*NPOS + 255)/256, 256, 0, stream>>>(pred, stats, uid, iid,
                                                        S0, S1, updated, contrib);
  k_scatter<<<(BATCH*NPOS + 255)/256, 256, 0, stream>>>(uid, iid, updated, out + 1);
  k_final  <<<1, 256, 0, stream>>>(contrib, fair, out);
}